// GraphSage_70265664963123
// MI455X (gfx1250) — compile-verified
//
#include <hip/hip_runtime.h>
#include <hip/hip_bf16.h>

#define N_NODES   50000
#define N_EDGES   1600000
#define FEATS     128          // in feats == hidden
#define KDIM      256          // concat(self, neigh)
#define DEG_PAD   50176        // 50000 rounded up

typedef float v2f __attribute__((ext_vector_type(2)));
typedef float v8f __attribute__((ext_vector_type(8)));

// ---------------------------------------------------------------- utilities
__global__ void fill_zero_kernel(float* __restrict__ p, int n) {
    int i = blockIdx.x * blockDim.x + threadIdx.x;
    if (i < n) p[i] = 0.0f;
}

__global__ void degree_kernel(const int* __restrict__ dst,
                              float* __restrict__ deg, int nEdges) {
    int e = blockIdx.x * blockDim.x + threadIdx.x;
    if (e < nEdges) {
        __hip_atomic_fetch_add(&deg[dst[e]], 1.0f,
                               __ATOMIC_RELAXED, __HIP_MEMORY_SCOPE_AGENT);
    }
}

__global__ void invdeg_kernel(float* __restrict__ deg, int n) {
    int i = blockIdx.x * blockDim.x + threadIdx.x;
    if (i < n) deg[i] = 1.0f / fmaxf(deg[i], 1.0f);
}

// ------------------------------------------------- edge scatter (mean numerator)
// one wave per edge; each lane moves 4 contiguous floats (32*4 = 128 feats)
__global__ void scatter_add_kernel(const float* __restrict__ h,
                                   const int* __restrict__ src,
                                   const int* __restrict__ dst,
                                   float* __restrict__ hn, int nEdges) {
    int e = blockIdx.x * (blockDim.x >> 5) + (threadIdx.x >> 5);
    if (e >= nEdges) return;
    int lane = threadIdx.x & 31;
    int s = src[e];
    int d = dst[e];
    const float4 v = *(const float4*)(h + (size_t)s * FEATS + lane * 4);
    float* p = hn + (size_t)d * FEATS + lane * 4;
    __hip_atomic_fetch_add(p + 0, v.x, __ATOMIC_RELAXED, __HIP_MEMORY_SCOPE_AGENT);
    __hip_atomic_fetch_add(p + 1, v.y, __ATOMIC_RELAXED, __HIP_MEMORY_SCOPE_AGENT);
    __hip_atomic_fetch_add(p + 2, v.z, __ATOMIC_RELAXED, __HIP_MEMORY_SCOPE_AGENT);
    __hip_atomic_fetch_add(p + 3, v.w, __ATOMIC_RELAXED, __HIP_MEMORY_SCOPE_AGENT);
}

// ------------------------------------------------------------- WMMA GEMM layer
// out[32-row tile] = act( concat(h, invdeg*hn) @ W + b )
// block = 256 threads = 8 wave32; wave w -> output cols [16w, 16w+16),
// two 16x16 accumulators per wave (rows 0-15 / 16-31) sharing B fragments.
// A tile in LDS, stride 260 (260 % 64 == 4 -> conflict-free 16-lane row reads,
// 260 % 4 == 0 -> float4-aligned fill). FOUT/ACT are compile-time so all W
// loads become immediate-offset global_load_b32 off one base pointer.
#define AS_STRIDE 260
#define ROWS_PB   32

template<int FOUT, int ACT>
__global__ __launch_bounds__(256)
void sage_gemm_kernel(const float* __restrict__ h,      // [N,128]
                      const float* __restrict__ hn,     // [N,128] sums
                      const float* __restrict__ invdeg, // [N]
                      const float* __restrict__ W,      // [256,FOUT]
                      const float* __restrict__ bias,   // [FOUT]
                      float* __restrict__ out) {        // [N,FOUT]
    __shared__ __align__(16) float As[ROWS_PB * AS_STRIDE];

    const int tid  = threadIdx.x;
    const int lane = tid & 31;
    const int wave = tid >> 5;
    const int row0 = blockIdx.x * ROWS_PB;

    // ---- cooperative A-tile load: 32 rows x 256 cols = 8192 floats.
    // thread t fills row t/8, 32 cols at (t%8)*32 (8 float4 stores).
    {
        const int r  = tid >> 3;          // 0..31
        const int c0 = (tid & 7) * 32;    // 0..224, span stays in one half
        const int nd = row0 + r;
        float4* dstp = (float4*)&As[r * AS_STRIDE + c0];
        if (nd < N_NODES) {
            if (c0 < FEATS) {             // self half
                const float4* sp = (const float4*)(h + (size_t)nd * FEATS + c0);
                #pragma unroll
                for (int j = 0; j < 8; ++j) dstp[j] = sp[j];
            } else {                      // neighbor half, scaled by 1/deg
                const float s = invdeg[nd];
                const float4* sp = (const float4*)(hn + (size_t)nd * FEATS + (c0 - FEATS));
                #pragma unroll
                for (int j = 0; j < 8; ++j) {
                    float4 v = sp[j];
                    v.x *= s; v.y *= s; v.z *= s; v.w *= s;
                    dstp[j] = v;
                }
            }
        } else {                          // tail rows of last block: zeros
            const float4 z = {0.f, 0.f, 0.f, 0.f};
            #pragma unroll
            for (int j = 0; j < 8; ++j) dstp[j] = z;
        }
    }
    __syncthreads();

    const int col0 = wave * 16;
    if (col0 >= FOUT) return;             // FOUT==64: waves 4..7 idle after load

    const int m    = lane & 15;           // A row (within tile) / B,C col
    const int ksel = (lane >> 4) << 1;    // 0 for lanes 0-15, 2 for 16-31

    // base pointers: all loop accesses become compile-time immediate offsets
    const float* ap0 = &As[m * AS_STRIDE + ksel];
    const float* ap1 = &As[(m + 16) * AS_STRIDE + ksel];
    const float* wp0 = W + ksel * FOUT + col0 + m;   // row k+ksel
    const float* wp1 = wp0 + FOUT;                   // row k+ksel+1

    v8f acc0 = {};
    v8f acc1 = {};
    #pragma unroll
    for (int k = 0; k < KDIM; k += 4) {
        const v2f a0 = *(const v2f*)(ap0 + k);       // ds_load_b64, imm offset
        const v2f a1 = *(const v2f*)(ap1 + k);
        v2f bf;
        bf.x = wp0[k * FOUT];                        // global_load_b32, imm offset
        bf.y = wp1[k * FOUT];
        acc0 = __builtin_amdgcn_wmma_f32_16x16x4_f32(
                   false, a0, false, bf, (short)0, acc0, false, false);
        acc1 = __builtin_amdgcn_wmma_f32_16x16x4_f32(
                   false, a1, false, bf, (short)0, acc1, false, false);
    }

    // ---- store: VGPR r -> row (r + 8*(lane>=16)), col = col0 + (lane&15)
    const int n     = col0 + m;
    const int rbase = row0 + ((lane >> 4) << 3);
    const float bv  = bias[n];
    #pragma unroll
    for (int r = 0; r < 8; ++r) {
        {   // tile 0: rows rbase + r
            const int row = rbase + r;
            float v = acc0[r] + bv;
            if (ACT) v = fmaxf(v, 0.0f);
            if (row < N_NODES) out[(size_t)row * FOUT + n] = v;
        }
        {   // tile 1: rows rbase + 16 + r
            const int row = rbase + 16 + r;
            float v = acc1[r] + bv;
            if (ACT) v = fmaxf(v, 0.0f);
            if (row < N_NODES) out[(size_t)row * FOUT + n] = v;
        }
    }
}

// ------------------------------------------------------------------ launcher
extern "C" void kernel_launch(void* const* d_in, const int* in_sizes, int n_in,
                              void* d_out, int out_size, void* d_ws, size_t ws_size,
                              hipStream_t stream) {
    const float* feat = (const float*)d_in[0];
    const int*   src  = (const int*)  d_in[1];
    const int*   dst  = (const int*)  d_in[2];
    const float* W0   = (const float*)d_in[3];
    const float* b0   = (const float*)d_in[4];
    const float* W1   = (const float*)d_in[5];
    const float* b1   = (const float*)d_in[6];
    const float* W2   = (const float*)d_in[7];
    const float* b2   = (const float*)d_in[8];
    float* out = (float*)d_out;

    float* ws  = (float*)d_ws;
    float* deg = ws;                                   // DEG_PAD floats
    float* hn  = deg + DEG_PAD;                        // 50000*128
    float* h1  = hn + (size_t)N_NODES * FEATS;         // 50000*128
    float* h2  = h1 + (size_t)N_NODES * FEATS;         // 50000*128

    const int nh = N_NODES * FEATS;                    // 6,400,000

    // degrees -> inverse degrees (computed once, reused by all 3 layers)
    fill_zero_kernel<<<(N_NODES + 255) / 256, 256, 0, stream>>>(deg, N_NODES);
    degree_kernel<<<(N_EDGES + 255) / 256, 256, 0, stream>>>(dst, deg, N_EDGES);
    invdeg_kernel<<<(N_NODES + 255) / 256, 256, 0, stream>>>(deg, N_NODES);

    const int scatterBlocks = (N_EDGES + 7) / 8;       // 8 waves (edges) / block
    const int gemmBlocks    = (N_NODES + ROWS_PB - 1) / ROWS_PB;  // 1563

    // ---- layer 0: feat -> h1 (relu)
    fill_zero_kernel<<<(nh + 255) / 256, 256, 0, stream>>>(hn, nh);
    scatter_add_kernel<<<scatterBlocks, 256, 0, stream>>>(feat, src, dst, hn, N_EDGES);
    sage_gemm_kernel<128, 1><<<gemmBlocks, 256, 0, stream>>>(feat, hn, deg, W0, b0, h1);

    // ---- layer 1: h1 -> h2 (relu)
    fill_zero_kernel<<<(nh + 255) / 256, 256, 0, stream>>>(hn, nh);
    scatter_add_kernel<<<scatterBlocks, 256, 0, stream>>>(h1, src, dst, hn, N_EDGES);
    sage_gemm_kernel<128, 1><<<gemmBlocks, 256, 0, stream>>>(h1, hn, deg, W1, b1, h2);

    // ---- layer 2: h2 -> out (no act, FOUT=64)
    fill_zero_kernel<<<(nh + 255) / 256, 256, 0, stream>>>(hn, nh);
    scatter_add_kernel<<<scatterBlocks, 256, 0, stream>>>(h2, src, dst, hn, N_EDGES);
    sage_gemm_kernel<64, 0><<<gemmBlocks, 256, 0, stream>>>(h2, hn, deg, W2, b2, out);
}